// StandGCN2_22428319219737
// MI455X (gfx1250) — compile-verified
//
#include <hip/hip_runtime.h>
#include <hip/hip_bf16.h>

// ---------------------------------------------------------------------------
// 2-layer GCN (PyG GCNConv semantics) for MI455X / gfx1250.
//   h1 = relu( Ahat @ (x @ W1) + b1 )        Ahat = D^-1/2 (A+I) D^-1/2
//   out = Ahat @ (h1 @ W2) + b2
// GEMMs use V_WMMA_F32_16X16X4_F32 (full fp32 path, matches reference
// precision). Aggregation = gather + f32 global atomics; the ~103MB working
// set fits entirely in the 192MB L2, so the scatter phase runs at L2 speed.
// ---------------------------------------------------------------------------

typedef float v2f __attribute__((ext_vector_type(2)));
typedef float v8f __attribute__((ext_vector_type(8)));

// ---------------- degree / normalization ----------------------------------

__global__ void deg_init(float* __restrict__ deg, int n) {
    int i = blockIdx.x * blockDim.x + threadIdx.x;
    if (i < n) deg[i] = 1.0f;                      // self-loop contributes 1
}

__global__ void deg_count(const int* __restrict__ dst, float* __restrict__ deg, int e) {
    int i = blockIdx.x * blockDim.x + threadIdx.x;
    if (i < e) atomicAdd(&deg[dst[i]], 1.0f);
}

__global__ void deg_to_dinv(float* __restrict__ deg, int n) {
    int i = blockIdx.x * blockDim.x + threadIdx.x;
    if (i < n) deg[i] = rsqrtf(deg[i]);            // deg >= 1 always (self-loops)
}

// ---------------- fp32 WMMA GEMM: C[nrows x NOUT] = X[nrows x 128] @ W -----
// One wave computes a 16-row x 64-col slab (4 accumulators share one A frag).
// A frag (16x4 f32): lanes 0-15 -> K = {k,k+1}, lanes 16-31 -> K = {k+2,k+3}.
// B frag (4x16 f32): VGPR v -> K row (v + 2*(lane>=16)), N = lane&15.
// C/D (16x16 f32):  VGPR v -> M = v + 8*(lane>=16), N = lane&15.

template <int NOUT>
__launch_bounds__(256)
__global__ void gemm_wmma_f32(const float* __restrict__ X,
                              const float* __restrict__ W,
                              float* __restrict__ C, int nrows) {
    static_assert(NOUT % 64 == 0, "NOUT must be a multiple of 64");
    constexpr int KDIM    = 128;
    constexpr int CTILES  = 4;               // 4 * 16 = 64 cols per wave
    constexpr int CGROUPS = NOUT / 64;

    const int gw   = (blockIdx.x * blockDim.x + threadIdx.x) >> 5;
    const int lane = threadIdx.x & 31;
    const int total_waves = (nrows / 16) * CGROUPS;
    if (gw >= total_waves) return;           // whole-wave exit: EXEC stays all-1s

    const int r0 = (gw / CGROUPS) * 16;
    const int c0 = (gw % CGROUPS) * 64;
    const int m  = lane & 15;
    const int kh = (lane >> 4) << 1;         // 0 for lanes 0-15, 2 for 16-31

    v8f acc[CTILES] = {};
    const float* xrow = X + (size_t)(r0 + m) * KDIM;

#pragma unroll 4
    for (int k = 0; k < KDIM; k += 4) {
        const v2f a = *(const v2f*)(xrow + k + kh);   // (k+kh) even -> 8B aligned
#pragma unroll
        for (int t = 0; t < CTILES; ++t) {
            const int n = c0 + t * 16 + m;
            v2f b;
            b.x = W[(size_t)(k + kh + 0) * NOUT + n];
            b.y = W[(size_t)(k + kh + 1) * NOUT + n];
            acc[t] = __builtin_amdgcn_wmma_f32_16x16x4_f32(
                /*neg_a=*/false, a, /*neg_b=*/false, b,
                /*c_mod=*/(short)0, acc[t], /*reuse_a=*/false, /*reuse_b=*/false);
        }
    }

    const int rbase = r0 + ((lane >> 4) << 3);        // +8 for upper half lanes
#pragma unroll
    for (int t = 0; t < CTILES; ++t) {
        const int n = c0 + t * 16 + m;
#pragma unroll
        for (int v = 0; v < 8; ++v)
            C[(size_t)(rbase + v) * NOUT + n] = acc[t][v];
    }
}

// ---------------- self-loop + bias init: agg = H * dinv^2 + b --------------

template <int F>
__global__ void init_agg(const float* __restrict__ H, const float* __restrict__ dinv,
                         const float* __restrict__ bias, float* __restrict__ agg, int n) {
    int idx = blockIdx.x * blockDim.x + threadIdx.x;
    if (idx >= n * F) return;
    const int row = idx / F;
    const int f   = idx - row * F;
    const float w = dinv[row];
    agg[idx] = H[idx] * w * w + bias[f];
}

// ---------------- edge scatter: agg[dst] += H[src] * dinv[s]*dinv[d] -------
// F=128: one edge per wave (32 float4 chunks). F=64: two edges per wave.

template <int F>
__launch_bounds__(256)
__global__ void edge_scatter(const float* __restrict__ H,
                             const int* __restrict__ src, const int* __restrict__ dst,
                             const float* __restrict__ dinv,
                             float* __restrict__ agg, int nedges) {
    constexpr int CH  = F / 4;               // float4 chunks per row
    constexpr int EPW = 32 / CH;             // edges per wave (1 or 2)
    const int wid  = (blockIdx.x * blockDim.x + threadIdx.x) >> 5;
    const int lane = threadIdx.x & 31;
    const int e  = wid * EPW + (EPW > 1 ? lane / CH : 0);
    const int cl = (EPW > 1) ? (lane & (CH - 1)) : lane;
    if (e >= nedges) return;

    const int s = src[e];
    const int d = dst[e];
    const float w = dinv[s] * dinv[d];
    const float4 v = ((const float4*)(H + (size_t)s * F))[cl];
    float* arow = agg + (size_t)d * F + cl * 4;
    atomicAdd(arow + 0, v.x * w);
    atomicAdd(arow + 1, v.y * w);
    atomicAdd(arow + 2, v.z * w);
    atomicAdd(arow + 3, v.w * w);
}

// ---------------- relu copy ------------------------------------------------

__global__ void relu_copy(const float* __restrict__ in, float* __restrict__ out, int total) {
    int idx = blockIdx.x * blockDim.x + threadIdx.x;
    if (idx < total) out[idx] = fmaxf(in[idx], 0.0f);
}

// ---------------------------------------------------------------------------

extern "C" void kernel_launch(void* const* d_in, const int* in_sizes, int n_in,
                              void* d_out, int out_size, void* d_ws, size_t ws_size,
                              hipStream_t stream) {
    (void)n_in; (void)out_size; (void)ws_size;

    const float* x  = (const float*)d_in[0];
    const int*   ei = (const int*)d_in[1];
    const float* W1 = (const float*)d_in[2];
    const float* b1 = (const float*)d_in[3];
    const float* W2 = (const float*)d_in[4];
    const float* b2 = (const float*)d_in[5];
    float* out = (float*)d_out;

    const int NFEAT  = 128;
    const int NHID   = 128;
    const int NCLASS = 64;
    const int N = in_sizes[0] / NFEAT;        // 100000 (multiple of 16)
    const int E = in_sizes[1] / 2;            // 600000
    const int* src = ei;
    const int* dst = ei + E;

    // workspace layout: dinv[N] | H[N x 128] | AGG[N x 128]
    char* ws = (char*)d_ws;
    float* dinv = (float*)ws;
    size_t off = (((size_t)N * 4) + 255) & ~(size_t)255;
    float* H = (float*)(ws + off);
    off += (size_t)N * 128 * 4;
    float* AGG = (float*)(ws + off);

    const int BLK = 256;
    auto cdiv = [](long long a, long long b) { return (int)((a + b - 1) / b); };

    // normalization
    deg_init<<<cdiv(N, BLK), BLK, 0, stream>>>(dinv, N);
    deg_count<<<cdiv(E, BLK), BLK, 0, stream>>>(dst, dinv, E);
    deg_to_dinv<<<cdiv(N, BLK), BLK, 0, stream>>>(dinv, N);

    // ---- layer 1: H = x @ W1 ; AGG = Ahat H + b1 ; H = relu(AGG) ----
    {
        const int waves = (N / 16) * (NHID / 64);
        gemm_wmma_f32<NHID><<<cdiv((long long)waves * 32, BLK), BLK, 0, stream>>>(x, W1, H, N);
        init_agg<NHID><<<cdiv((long long)N * NHID, BLK), BLK, 0, stream>>>(H, dinv, b1, AGG, N);
        edge_scatter<NHID><<<cdiv((long long)E * 32, BLK), BLK, 0, stream>>>(H, src, dst, dinv, AGG, E);
        relu_copy<<<cdiv((long long)N * NHID, BLK), BLK, 0, stream>>>(AGG, H, N * NHID);
    }

    // ---- layer 2: AGG = H @ W2 ; out = Ahat AGG + b2 ----
    {
        const int waves = (N / 16) * (NCLASS / 64);
        gemm_wmma_f32<NCLASS><<<cdiv((long long)waves * 32, BLK), BLK, 0, stream>>>(H, W2, AGG, N);
        init_agg<NCLASS><<<cdiv((long long)N * NCLASS, BLK), BLK, 0, stream>>>(AGG, dinv, b2, out, N);
        const int ewaves = (E + 1) / 2;       // 2 edges per wave at F=64
        edge_scatter<NCLASS><<<cdiv((long long)ewaves * 32, BLK), BLK, 0, stream>>>(AGG, src, dst, dinv, out, E);
    }
}